// ContextBaseTailAttention_27625229648477
// MI455X (gfx1250) — compile-verified
//
#include <hip/hip_runtime.h>
#include <hip/hip_bf16.h>

typedef __attribute__((ext_vector_type(2))) float    v2f;
typedef __attribute__((ext_vector_type(8))) float    v8f;
typedef __attribute__((ext_vector_type(4))) unsigned v4u;
typedef __attribute__((ext_vector_type(8))) unsigned v8u;

#define B_DIM 64
#define L_DIM 512
#define H_DIM 1024
#define LC    16                 // rows per TDM chunk (16*1024*4B = 64 KB)
#define NCHUNK (L_DIM / LC)      // 32

// LDS byte offset of a static __shared__ object (generic -> AS3 addrspacecast).
__device__ __forceinline__ unsigned lds_offset_of(const void* p) {
    return (unsigned)(unsigned long long)(__attribute__((address_space(3))) const char*)p;
}

// Issue one TDM 2-D tile load: LC x H_DIM f32 rows, row stride H_DIM, into LDS.
__device__ __forceinline__ void tdm_issue_chunk(const float* gsrc, unsigned ldsOff) {
    const unsigned long long ga = (unsigned long long)gsrc;
    v4u g0;
    g0.x = 1u;                                    // count=1, user mode
    g0.y = ldsOff;                                // lds_addr
    g0.z = (unsigned)ga;                          // global_addr[31:0]
    g0.w = (unsigned)(ga >> 32) | (2u << 30);     // global_addr[56:32] | type=2
    v8u g1;
    g1[0] = (2u << 16);                           // data_size=4B, no wg mask
    g1[1] = ((unsigned)H_DIM & 0xffffu) << 16;    // tensor_dim0 lo16
    g1[2] = (((unsigned)L_DIM & 0xffffu) << 16) | ((unsigned)H_DIM >> 16);
    g1[3] = (((unsigned)H_DIM & 0xffffu) << 16) | ((unsigned)L_DIM >> 16);
                                                  // tile_dim0 | tensor_dim1 hi
    g1[4] = (unsigned)LC;                         // tile_dim1 (rows)
    g1[5] = (unsigned)H_DIM;                      // tensor_dim0_stride lo32
    g1[6] = 0u;
    g1[7] = 0u;
    asm volatile("tensor_load_to_lds %0, %1" :: "s"(g0), "s"(g1) : "memory");
}

// ---------------------------------------------------------------------------
// Kernel A: G = h_t (64x1024) @ W^T (1024x1024): G[m,n] = sum_k h_t[m,k]*W[n,k]
// One wave32 per 16x64 output strip (4 N-tiles share one A fragment) using
// V_WMMA_F32_16X16X4_F32 along K.
// ---------------------------------------------------------------------------
__global__ __launch_bounds__(128)
void gemm_ht_wt_wmma(const float* __restrict__ h_t,
                     const float* __restrict__ W,
                     float* __restrict__ G) {
    const int wid  = blockIdx.x * 4 + (threadIdx.x >> 5); // 0..63 waves total
    const int tm   = wid >> 4;                            // 0..3 (M tile)
    const int tn0  = (wid & 15) * 4;                      // first of 4 N tiles
    const int lane = threadIdx.x & 31;
    const int lrow = lane & 15;
    const int lhi  = lane >> 4;

    const float* aPtr = h_t + (size_t)(tm * 16 + lrow) * H_DIM + 2 * lhi;
    const float* bPtr[4];
#pragma unroll
    for (int q = 0; q < 4; ++q)
        bPtr[q] = W + (size_t)((tn0 + q) * 16 + lrow) * H_DIM + 2 * lhi;

    v8f acc[4] = {v8f{}, v8f{}, v8f{}, v8f{}};
#pragma unroll 4
    for (int k0 = 0; k0 < H_DIM; k0 += 4) {
        v2f a = *(const v2f*)(aPtr + k0);
#pragma unroll
        for (int q = 0; q < 4; ++q) {
            v2f bq = *(const v2f*)(bPtr[q] + k0);
            acc[q] = __builtin_amdgcn_wmma_f32_16x16x4_f32(
                false, a, false, bq, (short)0, acc[q], false, false);
        }
    }

#pragma unroll
    for (int q = 0; q < 4; ++q) {
        float* outp = G + (size_t)(tm * 16 + 8 * lhi) * H_DIM + (tn0 + q) * 16 + lrow;
#pragma unroll
        for (int r = 0; r < 8; ++r) outp[(size_t)r * H_DIM] = acc[q][r];
    }
}

// ---------------------------------------------------------------------------
// Kernel B: per-batch fused scores -> online softmax -> weighted sum -> blend.
// cntx[b] streamed through LDS exactly once by the Tensor Data Mover with
// double-buffered chunks: DMA of chunk i+1 overlaps compute on chunk i
// (TENSORcnt completes in order, so s_wait_tensorcnt(1) pins chunk i done).
// ---------------------------------------------------------------------------
__global__ __launch_bounds__(256)
void fused_tail_attention(const float* __restrict__ h_t,
                          const float* __restrict__ cntx,
                          const float* __restrict__ G,
                          const float* __restrict__ mult_p,
                          float* __restrict__ out) {
    __shared__ float chunk[2][LC * H_DIM];  // 2 x 64 KB TDM destinations
    __shared__ float g_s[H_DIM];            // 4 KB
    __shared__ float sc[LC];

    const int b    = blockIdx.x;
    const int t    = threadIdx.x;
    const int wave = t >> 5;
    const int lane = t & 31;

    for (int i = t; i < H_DIM; i += 256) g_s[i] = G[(size_t)b * H_DIM + i];

    const float*   cb = cntx + (size_t)b * L_DIM * H_DIM;
    const unsigned ldsBuf[2] = {lds_offset_of(chunk[0]), lds_offset_of(chunk[1])};

    if (wave == 0) tdm_issue_chunk(cb, ldsBuf[0]);   // prologue: chunk 0 DMA

    float m_run  = -3.4e38f;
    float s_run  = 0.0f;
    float acc[4] = {0.f, 0.f, 0.f, 0.f};   // h = t + 256*j

    for (int ci = 0; ci < NCHUNK; ++ci) {
        __syncthreads();  // readers of buf[(ci+1)&1] from iter ci-1 are done

        if (wave == 0) {
            if (ci + 1 < NCHUNK) {
                tdm_issue_chunk(cb + (size_t)(ci + 1) * LC * H_DIM,
                                ldsBuf[(ci + 1) & 1]);
                __builtin_amdgcn_s_wait_tensorcnt(1);  // chunk ci complete
            } else {
                __builtin_amdgcn_s_wait_tensorcnt(0);
            }
        }
        __syncthreads();

        const float* buf = chunk[ci & 1];

        // ---- scores: 8 waves x 2 rows, wave32 shuffle reduction ----
#pragma unroll
        for (int lr = wave; lr < LC; lr += 8) {
            const float* row = buf + lr * H_DIM;
            float p = 0.0f;
#pragma unroll
            for (int h = lane; h < H_DIM; h += 32) p = fmaf(row[h], g_s[h], p);
            for (int off = 16; off > 0; off >>= 1) p += __shfl_down(p, off, 32);
            if (lane == 0) sc[lr] = p;
        }
        __syncthreads();

        // ---- online softmax update (redundant scalar state per thread) ----
        float cmax = sc[0];
#pragma unroll
        for (int l = 1; l < LC; ++l) cmax = fmaxf(cmax, sc[l]);
        const float m_new = fmaxf(m_run, cmax);
        const float scale = __expf(m_run - m_new);
        s_run *= scale;
#pragma unroll
        for (int j = 0; j < 4; ++j) acc[j] *= scale;

#pragma unroll
        for (int l = 0; l < LC; ++l) {
            const float p = __expf(sc[l] - m_new);
            s_run += p;
            const float* row = buf + l * H_DIM;
#pragma unroll
            for (int j = 0; j < 4; ++j) acc[j] = fmaf(p, row[t + 256 * j], acc[j]);
        }
        m_run = m_new;
    }

    const float p   = mult_p[0];
    const float q   = 1.0f - p;
    const float inv = 1.0f / s_run;
#pragma unroll
    for (int j = 0; j < 4; ++j) {
        const int h = t + 256 * j;
        out[(size_t)b * H_DIM + h] =
            p * h_t[(size_t)b * H_DIM + h] + q * (acc[j] * inv);
    }
}

// ---------------------------------------------------------------------------
extern "C" void kernel_launch(void* const* d_in, const int* in_sizes, int n_in,
                              void* d_out, int out_size, void* d_ws, size_t ws_size,
                              hipStream_t stream) {
    const float* h_t    = (const float*)d_in[0];
    const float* cntx   = (const float*)d_in[1];
    const float* W      = (const float*)d_in[2];
    const float* mult_p = (const float*)d_in[3];
    float*       out    = (float*)d_out;
    float*       G      = (float*)d_ws;   // B_DIM * H_DIM floats (256 KB)

    // Kernel A: 64 waves, each a 16x64 WMMA strip; 4 waves/block.
    gemm_ht_wt_wmma<<<dim3(16), dim3(128), 0, stream>>>(h_t, W, G);

    // Kernel B: one workgroup per batch row; double-buffered TDM streaming.
    fused_tail_attention<<<dim3(B_DIM), dim3(256), 0, stream>>>(
        h_t, cntx, G, mult_p, out);
}